// OriginalMalConv_36928128811589
// MI455X (gfx1250) — compile-verified
//
#include <hip/hip_runtime.h>
#include <math.h>

typedef __attribute__((ext_vector_type(2))) float v2f;
typedef __attribute__((ext_vector_type(4))) float v4f;
typedef __attribute__((ext_vector_type(8))) float v8f;

#define KCONV 500
#define CIN   8
#define COUT  128
#define NTOT  256                 // fused conv1+conv2 output columns
#define BATCH 8
#define LLEN  (1 << 20)
#define TWIN  2097                // (L-K)/K + 1
#define MROWS (BATCH * TWIN)      // 16776
#define KFLAT (KCONV * CIN)       // 4000
#define KPAIRS (KFLAT / 2)        // 2000

#define MT    64                  // M rows per block (4 subtiles of 16)
#define MSUB  4                   // M subtiles per wave
#define KC    160                 // kflat per LDS chunk (multiple of 8)
#define PAIRS (KC / 2)            // 80 float-pairs per chunk
#define NCHUNK (KFLAT / KC)       // 25
#define TOKS_PER_CHUNK (KC / CIN) // 20 tokens per row per chunk
#define KSTEPS (KC / 4)           // 40 wmma k-steps per chunk
#define ASTRIDE 80                // v2f stride per pair: 64 rows + 16 pad
                                  // -> h=1 half lands on banks 32..63 (no conflict)
#define SMEM_BYTES (PAIRS * ASTRIDE * 8)   // 51.2 KB (Cbuf 16 KB aliases this)

// ---------------------------------------------------------------------------
// Kernel 1: repack w1 (o,e,k) and w2 into Bp[kpair][256][2] so the WMMA
// B-fragment load is a single coalesced b64 per lane. kflat = 8*k + e.
// ---------------------------------------------------------------------------
__global__ void repack_w(const float* __restrict__ w1,
                         const float* __restrict__ w2,
                         float* __restrict__ Bp) {
  int idx = blockIdx.x * blockDim.x + threadIdx.x;   // over KPAIRS*NTOT
  if (idx >= KPAIRS * NTOT) return;
  int pg = idx / NTOT;
  int o  = idx % NTOT;
  int kf = 2 * pg;
  int e  = kf & 7;       // even
  int k  = kf >> 3;
  const float* w = (o < COUT) ? (w1 + (size_t)o * KFLAT)
                              : (w2 + (size_t)(o - COUT) * KFLAT);
  v2f val;
  val.x = w[e * KCONV + k];
  val.y = w[(e + 1) * KCONV + k];
  ((v2f*)Bp)[(size_t)pg * NTOT + o] = val;
}

// ---------------------------------------------------------------------------
// Kernel 2: zero the pooled max buffer each launch (gated values are >= 0,
// so 0 is the identity for max and int-compare == float-compare).
// ---------------------------------------------------------------------------
__global__ void zero_pooled(float* __restrict__ pooled) {
  for (int j = threadIdx.x; j < BATCH * COUT; j += blockDim.x) pooled[j] = 0.0f;
}

// ---------------------------------------------------------------------------
// Kernel 3: fused embed -> window GEMM (fp32 WMMA, 64x256 block tile)
//           -> bias -> relu*sigmoid -> global max-pool (int atomicMax).
// 8 wave32 per block; wave owns 4 M-subtiles x 2 N-tiles (8 v8f accums).
// ---------------------------------------------------------------------------
__global__ __launch_bounds__(256)
void gemm_gate_pool(const int*   __restrict__ x,
                    const float* __restrict__ emb,
                    const float* __restrict__ b1,
                    const float* __restrict__ b2,
                    const float* __restrict__ Bp,
                    float*       __restrict__ pooled) {
  __shared__ __align__(16) char smem[SMEM_BYTES];
  v2f*   As   = (v2f*)smem;            // during K loop: [pair][ASTRIDE rows]
  float* Cbuf = (float*)smem;          // after K loop:  [16][256] per subtile

  const int tid  = threadIdx.x;
  const int lane = tid & 31;
  const int wave = tid >> 5;
  const int r    = lane & 15;   // A row within 16 / B,C column within 16
  const int h    = lane >> 4;   // lane half -> K group / C row group
  const int tile = blockIdx.x;
  const int o0   = wave * 32 + r;
  const int o1   = o0 + 16;

  v8f acc00 = {}, acc01 = {};   // subtile 0, n-tile 0/1
  v8f acc10 = {}, acc11 = {};   // subtile 1
  v8f acc20 = {}, acc21 = {};   // subtile 2
  v8f acc30 = {}, acc31 = {};   // subtile 3
  const v2f* __restrict__ Bpv = (const v2f*)Bp;

  for (int c = 0; c < NCHUNK; ++c) {
    __syncthreads();  // previous chunk's readers are done
    // ---- stage A chunk: 64 rows x 20 tokens, embed on the fly ----
    for (int j = tid; j < MT * TOKS_PER_CHUNK; j += 256) {
      int rr = j & 63;          // row fastest -> stores spread across banks
      int kl = j >> 6;          // 0..19
      int k  = c * TOKS_PER_CHUNK + kl;
      int m  = tile * MT + rr;
      int tok = 256;            // padding row (zeros) for out-of-range rows
      if (m < MROWS) {
        int b = m / TWIN;
        int t = m % TWIN;
        tok = x[b * LLEN + t * KCONV + k];
      }
      const v4f* er = (const v4f*)(emb + tok * CIN);
      v4f e0 = er[0];
      v4f e1 = er[1];
      int pbase = 4 * kl;       // local kflat-pair base for this token
      v2f p0; p0.x = e0.x; p0.y = e0.y;
      v2f p1; p1.x = e0.z; p1.y = e0.w;
      v2f p2; p2.x = e1.x; p2.y = e1.y;
      v2f p3; p3.x = e1.z; p3.y = e1.w;
      As[(pbase + 0) * ASTRIDE + rr] = p0;
      As[(pbase + 1) * ASTRIDE + rr] = p1;
      As[(pbase + 2) * ASTRIDE + rr] = p2;
      As[(pbase + 3) * ASTRIDE + rr] = p3;
    }
    __syncthreads();
    // ---- WMMA over this chunk: 40 k-steps of 16x16x4 fp32, 8 wmma/step ----
    const v2f* __restrict__ Bc = Bpv + ((size_t)c * PAIRS + h) * NTOT;
    for (int kk = 0; kk < KSTEPS; ++kk) {
      int pb = (2 * kk + h) * ASTRIDE + r;   // conflict-free: h=1 -> +32 banks
      v2f a0 = As[pb];
      v2f a1 = As[pb + 16];
      v2f a2 = As[pb + 32];
      v2f a3 = As[pb + 48];
      const v2f* bp = Bc + (size_t)(2 * kk) * NTOT;
      v2f bb0 = bp[o0];
      v2f bb1 = bp[o1];
      acc00 = __builtin_amdgcn_wmma_f32_16x16x4_f32(false, a0, false, bb0, (short)0, acc00, false, false);
      acc01 = __builtin_amdgcn_wmma_f32_16x16x4_f32(false, a0, false, bb1, (short)0, acc01, false, false);
      acc10 = __builtin_amdgcn_wmma_f32_16x16x4_f32(false, a1, false, bb0, (short)0, acc10, false, false);
      acc11 = __builtin_amdgcn_wmma_f32_16x16x4_f32(false, a1, false, bb1, (short)0, acc11, false, false);
      acc20 = __builtin_amdgcn_wmma_f32_16x16x4_f32(false, a2, false, bb0, (short)0, acc20, false, false);
      acc21 = __builtin_amdgcn_wmma_f32_16x16x4_f32(false, a2, false, bb1, (short)0, acc21, false, false);
      acc30 = __builtin_amdgcn_wmma_f32_16x16x4_f32(false, a3, false, bb0, (short)0, acc30, false, false);
      acc31 = __builtin_amdgcn_wmma_f32_16x16x4_f32(false, a3, false, bb1, (short)0, acc31, false, false);
    }
  }

  // ---- per M-subtile: spill C (+bias) to LDS, gate, max-pool ----
  float bias0 = (o0 < COUT) ? b1[o0] : b2[o0 - COUT];
  float bias1 = (o1 < COUT) ? b1[o1] : b2[o1 - COUT];
  v8f s0[MSUB] = {acc00, acc10, acc20, acc30};
  v8f s1[MSUB] = {acc01, acc11, acc21, acc31};
#pragma unroll
  for (int ms = 0; ms < MSUB; ++ms) {
    __syncthreads();            // previous pass's readers done (Cbuf aliases As)
#pragma unroll
    for (int v = 0; v < 8; ++v) {
      int row = v + 8 * h;      // C layout: M = v + 8*(lane/16), N = lane%16
      Cbuf[row * NTOT + o0] = s0[ms][v] + bias0;
      Cbuf[row * NTOT + o1] = s1[ms][v] + bias1;
    }
    __syncthreads();
    for (int j = tid; j < 16 * COUT; j += 256) {
      int rr = j >> 7;
      int o  = j & 127;
      int m  = tile * MT + ms * 16 + rr;
      if (m < MROWS) {
        float c1 = Cbuf[rr * NTOT + o];
        float c2 = Cbuf[rr * NTOT + COUT + o];
        float relu = c1 > 0.0f ? c1 : 0.0f;
        float gate = relu / (1.0f + __expf(-c2));  // relu(c1)*sigmoid(c2) >= 0
        int b = m / TWIN;
        atomicMax((int*)&pooled[b * COUT + o], __float_as_int(gate));
      }
    }
  }
}

// ---------------------------------------------------------------------------
// Kernel 4: tiny dense head: h = relu(pooled @ wd1^T + bd1); out = sigmoid(..)
// ---------------------------------------------------------------------------
__global__ __launch_bounds__(128)
void head_kernel(const float* __restrict__ pooled,
                 const float* __restrict__ wd1, const float* __restrict__ bd1,
                 const float* __restrict__ wd2, const float* __restrict__ bd2,
                 float* __restrict__ out) {
  __shared__ float pbuf[BATCH * COUT];
  __shared__ float hred[COUT];
  int o = threadIdx.x;
  for (int j = o; j < BATCH * COUT; j += 128) pbuf[j] = pooled[j];
  __syncthreads();
  for (int b = 0; b < BATCH; ++b) {
    float s = bd1[o];
#pragma unroll 4
    for (int k = 0; k < COUT; ++k) s += pbuf[b * COUT + k] * wd1[o * COUT + k];
    float hv = s > 0.0f ? s : 0.0f;
    hred[o] = hv * wd2[o];
    __syncthreads();
    if (o == 0) {
      float t = bd2[0];
      for (int k = 0; k < COUT; ++k) t += hred[k];
      out[b] = 1.0f / (1.0f + __expf(-t));
    }
    __syncthreads();
  }
}

// ---------------------------------------------------------------------------
extern "C" void kernel_launch(void* const* d_in, const int* in_sizes, int n_in,
                              void* d_out, int out_size, void* d_ws, size_t ws_size,
                              hipStream_t stream) {
  const int*   x   = (const int*)d_in[0];
  const float* emb = (const float*)d_in[1];
  const float* w1  = (const float*)d_in[2];
  const float* b1  = (const float*)d_in[3];
  const float* w2  = (const float*)d_in[4];
  const float* b2  = (const float*)d_in[5];
  const float* wd1 = (const float*)d_in[6];
  const float* bd1 = (const float*)d_in[7];
  const float* wd2 = (const float*)d_in[8];
  const float* bd2 = (const float*)d_in[9];

  float* Bp     = (float*)d_ws;                       // 2000*256*2 f32 = 4 MB
  float* pooled = Bp + (size_t)KPAIRS * NTOT * 2;     // 1024 f32
  float* out    = (float*)d_out;                      // 8 f32

  int repackN = KPAIRS * NTOT;
  repack_w<<<(repackN + 255) / 256, 256, 0, stream>>>(w1, w2, Bp);
  zero_pooled<<<1, 256, 0, stream>>>(pooled);

  int tiles = (MROWS + MT - 1) / MT;                  // 263 blocks (~1/WGP)
  gemm_gate_pool<<<tiles, 256, 0, stream>>>(x, emb, b1, b2, Bp, pooled);

  head_kernel<<<1, 128, 0, stream>>>(pooled, wd1, bd1, wd2, bd2, out);
}